// gnnModel_6425271075225
// MI455X (gfx1250) — compile-verified
//
#include <hip/hip_runtime.h>

typedef __attribute__((ext_vector_type(2))) float v2f;
typedef __attribute__((ext_vector_type(8))) float v8f;

#define N_ROWS   8192
#define F_IN     1433
#define F_OUT    7
#define NPAD     16
#define K_WAVES  8            // waves per block splitting K
#define K_PER_W  (N_ROWS / K_WAVES)   // 1024

// ---------------------------------------------------------------------------
// Kernel 1: y = x @ W.T ; z_pad[i][j] = d_i * y[i][j] (j<7, 0 padded to 16),
//           dvec[i] = d_i = rsqrt(dig[i]).
// One wave per row; lanes stride over F_IN, butterfly-reduce 7 accumulators.
// ---------------------------------------------------------------------------
__global__ void __launch_bounds__(256)
zy_kernel(const float* __restrict__ x, const float* __restrict__ dig,
          const float* __restrict__ W, float* __restrict__ z_pad,
          float* __restrict__ dvec) {
    const int gtid = blockIdx.x * blockDim.x + threadIdx.x;
    const int row  = gtid >> 5;        // wave index == row
    const int lane = threadIdx.x & 31;

    const float* xr = x + (size_t)row * F_IN;

    float acc[F_OUT];
#pragma unroll
    for (int j = 0; j < F_OUT; ++j) acc[j] = 0.0f;

    for (int k = lane; k < F_IN; k += 32) {
        float xv = xr[k];
#pragma unroll
        for (int j = 0; j < F_OUT; ++j)
            acc[j] += xv * W[j * F_IN + k];
    }

    // butterfly reduction across the 32 lanes (all lanes end with the sum)
#pragma unroll
    for (int j = 0; j < F_OUT; ++j) {
#pragma unroll
        for (int off = 16; off > 0; off >>= 1)
            acc[j] += __shfl_xor(acc[j], off, 32);
    }

    const float d = rsqrtf(dig[row]);
    if (lane == 0) {
        dvec[row] = d;
        float* zr = z_pad + (size_t)row * NPAD;
#pragma unroll
        for (int j = 0; j < F_OUT; ++j) zr[j] = d * acc[j];
#pragma unroll
        for (int j = F_OUT; j < NPAD; ++j) zr[j] = 0.0f;
    }
}

// ---------------------------------------------------------------------------
// Kernel 2: out[i][n] = d_i * (A @ z_pad)[i][n] + 2*d_i*z_pad[i][n] + b[n]
// One block (8 waves) per 16-row tile of A. Each wave accumulates a 16x16
// f32 tile with V_WMMA_F32_16X16X4_F32 over a 1024-wide K slice; partial
// tiles are reduced across waves through LDS; fused scaled epilogue.
// ---------------------------------------------------------------------------
__global__ void __launch_bounds__(256)
spmm_wmma_kernel(const float* __restrict__ A, const float* __restrict__ z_pad,
                 const float* __restrict__ dvec, const float* __restrict__ bias,
                 float* __restrict__ out) {
    __shared__ float red[K_WAVES * 256];   // 8 partial 16x16 tiles (8 KB)

    const int tid  = threadIdx.x;
    const int wave = tid >> 5;             // 0..7  -> K slice
    const int lane = tid & 31;
    const int half = lane >> 4;            // 0: lanes 0-15, 1: lanes 16-31
    const int l16  = lane & 15;
    const int row0 = blockIdx.x * 16;

    const int k0 = wave * K_PER_W;

    // A fragment (16x4 f32): lane holds row M=l16; VGPR pair = K {h2, h2+1},
    // where h2 = 2*half. Contiguous 8B per lane -> global_load_b64.
    const float* aptr = A + (size_t)(row0 + l16) * N_ROWS + k0 + half * 2;
    // B fragment (4x16 f32): lane holds col N=l16; rows K {h2, h2+1}.
    const float* bptr = z_pad + (size_t)(k0 + half * 2) * NPAD + l16;

    v8f c = {};   // zero accumulator

    for (int kk = 0; kk < K_PER_W; kk += 64) {
        // pull the A stream ahead into cache (global_prefetch_b8)
        __builtin_prefetch(aptr + kk + 2048, 0, 0);
#pragma unroll
        for (int t = 0; t < 64; t += 4) {
            const int k = kk + t;
            v2f a = *(const v2f*)(aptr + k);
            v2f bb;
            bb.x = bptr[k * NPAD];
            bb.y = bptr[k * NPAD + NPAD];
            // D = A(16x4) * B(4x16) + C
            c = __builtin_amdgcn_wmma_f32_16x16x4_f32(
                    false, a, false, bb, (short)0, c, false, false);
        }
    }

    // stash this wave's partial C tile
#pragma unroll
    for (int v = 0; v < 8; ++v)
        red[wave * 256 + v * 32 + lane] = c[v];
    __syncthreads();

    // cross-wave reduction: each thread owns one C element (flat index = tid)
    float s = 0.0f;
#pragma unroll
    for (int w = 0; w < K_WAVES; ++w)
        s += red[w * 256 + tid];

    // C layout: VGPR v, lane L : M = (L<16 ? v : v+8), N = L&15
    const int vv = tid >> 5;
    const int ln = tid & 31;
    const int m  = (ln < 16) ? vv : vv + 8;
    const int n  = ln & 15;

    if (n < F_OUT) {
        const int i   = row0 + m;
        const float d = dvec[i];
        out[(size_t)i * F_OUT + n] =
            d * s + 2.0f * d * z_pad[(size_t)i * NPAD + n] + bias[n];
    }
}

// ---------------------------------------------------------------------------
extern "C" void kernel_launch(void* const* d_in, const int* in_sizes, int n_in,
                              void* d_out, int out_size, void* d_ws, size_t ws_size,
                              hipStream_t stream) {
    const float* x   = (const float*)d_in[0];   // [8192,1433]
    const float* dig = (const float*)d_in[1];   // [8192]
    const float* A   = (const float*)d_in[2];   // [8192,8192]
    const float* W   = (const float*)d_in[3];   // [7,1433]
    const float* b   = (const float*)d_in[4];   // [7]
    float* out = (float*)d_out;                 // [8192,7]

    float* z_pad = (float*)d_ws;                // 8192*16 f32 = 512 KB
    float* dvec  = z_pad + (size_t)N_ROWS * NPAD; // 8192 f32

    // y = x @ W.T fused with d-scaling -> z_pad
    zy_kernel<<<N_ROWS / 8, 256, 0, stream>>>(x, dig, W, z_pad, dvec);

    // out = d ⊙ (A @ z) + 2 d ⊙ z + b   (memory-bound single pass over A)
    spmm_wmma_kernel<<<N_ROWS / 16, 256, 0, stream>>>(A, z_pad, dvec, b, out);
}